// LTMModule_10033043603916
// MI455X (gfx1250) — compile-verified
//
#include <hip/hip_runtime.h>

#define N_SLOTS   65536
#define KEY_DIM   64
#define VAL_DIM   64
#define N_QUERIES 8192
#define TOPK      4
#define LR        1e-3f
#define MOM       0.9f
#define WD        1e-4f

#define NCHUNK      8
#define CHUNK_SLOTS (N_SLOTS / NCHUNK)    // 8192 slots per workgroup
#define STAGE_SLOTS 64                    // key rows per async stage (double buffered)
#define NSTAGES     (CHUNK_SLOTS / STAGE_SLOTS)   // 128
#define WAVES_WG    4
#define QT_WAVE     16                    // query rows per wave (WMMA M)
#define QT_WG       (WAVES_WG * QT_WAVE)  // 64 queries per workgroup
#define LDS_STRIDE  68                    // 64 + 4 pad: 16B-aligned rows, conflict-free

typedef float v2f __attribute__((ext_vector_type(2)));
typedef float v8f __attribute__((ext_vector_type(8)));
typedef int   v4i __attribute__((vector_size(16)));

typedef __attribute__((address_space(1))) v4i* glb_v4i_p;
typedef __attribute__((address_space(3))) v4i* lds_v4i_p;

// ---------------------------------------------------------------------------
// CDNA5 async global->LDS copy (ASYNCcnt tracked, no VGPR data round-trip)
// ---------------------------------------------------------------------------
__device__ __forceinline__ void async_copy16(const float* gsrc, float* ldst)
{
#if defined(__has_builtin) && __has_builtin(__builtin_amdgcn_global_load_async_to_lds_b128)
    // two-step casts: addrspacecast to AS-qualified pointer, then bitcast to int4*
    const __attribute__((address_space(1))) float* g1 =
        (const __attribute__((address_space(1))) float*)gsrc;
    __attribute__((address_space(3))) float* l3 =
        (__attribute__((address_space(3))) float*)ldst;
    __builtin_amdgcn_global_load_async_to_lds_b128((glb_v4i_p)g1, (lds_v4i_p)l3, 0, 0);
#else
    unsigned loff = (unsigned)(size_t)(void*)ldst;   // flat addr low 32 bits = LDS offset
    asm volatile("global_load_async_to_lds_b128 %0, %1, off"
                 :: "v"(loff), "v"(gsrc) : "memory");
#endif
}

__device__ __forceinline__ void wait_async0()
{
#if defined(__has_builtin) && __has_builtin(__builtin_amdgcn_s_wait_asynccnt)
    __builtin_amdgcn_s_wait_asynccnt(0);
#else
    asm volatile("s_wait_asynccnt 0x0" ::: "memory");
#endif
}

// ---------------------------------------------------------------------------
// Kernel 1: fused fp32 GEMM (V_WMMA_F32_16X16X4_F32) + per-chunk top-4,
//           double-buffered async-to-LDS key staging
// ---------------------------------------------------------------------------
__global__ __launch_bounds__(128) void topk_gemm(const float* __restrict__ Q,
                                                 const float* __restrict__ Keys,
                                                 float* __restrict__ pvals,
                                                 int* __restrict__ pidx)
{
    __shared__ float smem[2 * STAGE_SLOTS * LDS_STRIDE];  // 34816 B, reused for merge

    const int tid  = threadIdx.x;
    const int wave = tid >> 5;
    const int lane = tid & 31;
    const int half = lane >> 4;   // 0: K pair {0,1}, 1: K pair {2,3}
    const int l16  = lane & 15;   // A: row M, B: col N
    const int kb   = half * 2;

    const int qtile = blockIdx.x;  // 0..127
    const int chunk = blockIdx.y;  // 0..7
    const int qbase = qtile * QT_WG + wave * QT_WAVE;

    // ---- A fragments: 16 rows x 64 K, held in VGPRs for the whole kernel ----
    v2f a[16];
    {
        const float* qr = Q + (size_t)(qbase + l16) * KEY_DIM;
        #pragma unroll
        for (int kk = 0; kk < 16; ++kk)
            a[kk] = *(const v2f*)(qr + kk * 4 + kb);
    }

    // ---- per-lane running top-4 for each of the 8 C rows this lane holds ----
    float tv[8][4];
    int   ti[8][4];
    #pragma unroll
    for (int j = 0; j < 8; ++j)
        #pragma unroll
        for (int t = 0; t < 4; ++t) { tv[j][t] = -3.4e38f; ti[j][t] = -1; }

    const int slot0 = chunk * CHUNK_SLOTS;

    // prefetch stage 0 into buffer 0 (8 async b128 per thread = 16 KB total)
    {
        const float* src = Keys + (size_t)slot0 * KEY_DIM;
        #pragma unroll
        for (int i = 0; i < 8; ++i) {
            int flat = (i * 128 + tid) * 4;          // 0..4092, multiple of 4
            int r = flat >> 6, c = flat & 63;
            async_copy16(src + flat, &smem[r * LDS_STRIDE + c]);
        }
    }

    for (int s = 0; s < NSTAGES; ++s) {
        const int   cur = s & 1;
        float*      buf = &smem[cur * STAGE_SLOTS * LDS_STRIDE];

        wait_async0();        // own async copies for buf[cur] complete
        __syncthreads();      // everyone's copies complete; prev compute done

        if (s + 1 < NSTAGES) {                       // overlap next copy with WMMA
            float*       nbuf = &smem[(cur ^ 1) * STAGE_SLOTS * LDS_STRIDE];
            const float* src  = Keys + (size_t)(slot0 + (s + 1) * STAGE_SLOTS) * KEY_DIM;
            #pragma unroll
            for (int i = 0; i < 8; ++i) {
                int flat = (i * 128 + tid) * 4;
                int r = flat >> 6, c = flat & 63;
                async_copy16(src + flat, &nbuf[r * LDS_STRIDE + c]);
            }
        }

        // 4 tiles of 16 slots, processed in pairs -> 2 independent WMMA chains
        #pragma unroll
        for (int p = 0; p < 2; ++p) {
            v8f c0 = {0.f,0.f,0.f,0.f,0.f,0.f,0.f,0.f};
            v8f c1 = {0.f,0.f,0.f,0.f,0.f,0.f,0.f,0.f};
            const float* b0 = &buf[((2*p + 0) * 16 + l16) * LDS_STRIDE + kb];
            const float* b1 = &buf[((2*p + 1) * 16 + l16) * LDS_STRIDE + kb];
            #pragma unroll
            for (int kk = 0; kk < 16; ++kk) {
                v2f bb0 = *(const v2f*)(b0 + kk * 4);
                v2f bb1 = *(const v2f*)(b1 + kk * 4);
                c0 = __builtin_amdgcn_wmma_f32_16x16x4_f32(
                        false, a[kk], false, bb0, (short)0, c0, false, false);
                c1 = __builtin_amdgcn_wmma_f32_16x16x4_f32(
                        false, a[kk], false, bb1, (short)0, c1, false, false);
            }
            const int sbase = slot0 + s * STAGE_SLOTS + 2 * p * 16 + l16;
            #pragma unroll
            for (int j = 0; j < 8; ++j) {
                #pragma unroll
                for (int h = 0; h < 2; ++h) {
                    float v  = h ? c1[j] : c0[j];
                    int   sg = sbase + h * 16;
                    if (v > tv[j][3]) {
                        bool g0 = v > tv[j][0], g1 = v > tv[j][1], g2 = v > tv[j][2];
                        float v0 = tv[j][0], v1 = tv[j][1], v2 = tv[j][2];
                        int   i0 = ti[j][0], i1 = ti[j][1], i2 = ti[j][2];
                        tv[j][0] = g0 ? v  : v0;             ti[j][0] = g0 ? sg : i0;
                        tv[j][1] = g0 ? v0 : (g1 ? v : v1);  ti[j][1] = g0 ? i0 : (g1 ? sg : i1);
                        tv[j][2] = g1 ? v1 : (g2 ? v : v2);  ti[j][2] = g1 ? i1 : (g2 ? sg : i2);
                        tv[j][3] = g2 ? v2 : v;              ti[j][3] = g2 ? i2 : sg;
                    }
                }
            }
        }
    }

    __syncthreads();   // last compute done before reusing smem for the merge

    // ---- cross-lane merge via LDS: 16 lanes x 4 candidates per query row ----
    // vals at smem[0..4096), idx bits at smem[4096..8192)
    #pragma unroll
    for (int j = 0; j < 8; ++j) {
        int m  = j + half * 8;
        int ql = wave * QT_WAVE + m;
        #pragma unroll
        for (int t = 0; t < 4; ++t) {
            smem[ql * 64 + l16 * 4 + t]        = tv[j][t];
            smem[4096 + ql * 64 + l16 * 4 + t] = __int_as_float(ti[j][t]);
        }
    }
    __syncthreads();

    if (tid < QT_WG) {
        float bv[4] = {-3.4e38f, -3.4e38f, -3.4e38f, -3.4e38f};
        int   bi[4] = {-1, -1, -1, -1};
        for (int cidx = 0; cidx < 64; ++cidx) {
            float v = smem[tid * 64 + cidx];
            int   sl = __float_as_int(smem[4096 + tid * 64 + cidx]);
            if (v > bv[3]) {
                bool g0 = v > bv[0], g1 = v > bv[1], g2 = v > bv[2];
                float v0 = bv[0], v1 = bv[1], v2 = bv[2];
                int   i0 = bi[0], i1 = bi[1], i2 = bi[2];
                bv[0] = g0 ? v  : v0;            bi[0] = g0 ? sl : i0;
                bv[1] = g0 ? v0 : (g1 ? v : v1); bi[1] = g0 ? i0 : (g1 ? sl : i1);
                bv[2] = g1 ? v1 : (g2 ? v : v2); bi[2] = g1 ? i1 : (g2 ? sl : i2);
                bv[3] = g2 ? v2 : v;             bi[3] = g2 ? i2 : sl;
            }
        }
        int qg   = qtile * QT_WG + tid;
        int base = (qg * NCHUNK + chunk) * TOPK;
        #pragma unroll
        for (int t = 0; t < 4; ++t) { pvals[base + t] = bv[t]; pidx[base + t] = bi[t]; }
    }
}

// ---------------------------------------------------------------------------
// Kernel 2: merge per-chunk partial top-4 lists -> final indices
// ---------------------------------------------------------------------------
__global__ void merge_topk(const float* __restrict__ pvals, const int* __restrict__ pidx,
                           float* __restrict__ out_idx_f, int* __restrict__ fidx)
{
    int q = blockIdx.x * blockDim.x + threadIdx.x;
    if (q >= N_QUERIES) return;
    float bv[4] = {-3.4e38f, -3.4e38f, -3.4e38f, -3.4e38f};
    int   bi[4] = {-1, -1, -1, -1};
    const float* pv = pvals + (size_t)q * NCHUNK * TOPK;
    const int*   pi = pidx  + (size_t)q * NCHUNK * TOPK;
    for (int c = 0; c < NCHUNK * TOPK; ++c) {
        float v = pv[c];
        int   s = pi[c];
        if (v > bv[3]) {
            bool g0 = v > bv[0], g1 = v > bv[1], g2 = v > bv[2];
            float v0 = bv[0], v1 = bv[1], v2 = bv[2];
            int   i0 = bi[0], i1 = bi[1], i2 = bi[2];
            bv[0] = g0 ? v  : v0;            bi[0] = g0 ? s  : i0;
            bv[1] = g0 ? v0 : (g1 ? v : v1); bi[1] = g0 ? i0 : (g1 ? s : i1);
            bv[2] = g1 ? v1 : (g2 ? v : v2); bi[2] = g1 ? i1 : (g2 ? s : i2);
            bv[3] = g2 ? v2 : v;             bi[3] = g2 ? i2 : s;
        }
    }
    #pragma unroll
    for (int t = 0; t < TOPK; ++t) {
        out_idx_f[q * TOPK + t] = (float)bi[t];
        fidx[q * TOPK + t] = bi[t];
    }
}

// ---------------------------------------------------------------------------
// Kernel 3: retrieved = vals[idx]  (old vals)
// ---------------------------------------------------------------------------
__global__ void gather_vals(const int* __restrict__ fidx, const float* __restrict__ vals,
                            float* __restrict__ out_retr)
{
    int gid = blockIdx.x * blockDim.x + threadIdx.x;   // over N_QUERIES*TOPK*VAL_DIM
    int d = gid & (VAL_DIM - 1);
    int p = gid >> 6;
    int s = fidx[p];
    out_retr[gid] = vals[(size_t)s * VAL_DIM + d];
}

// ---------------------------------------------------------------------------
// Kernel 4: zero accumulators (harness poisons ws with 0xAA)
// ---------------------------------------------------------------------------
__global__ void zero_ws(float* __restrict__ sgrads, float* __restrict__ counts)
{
    const int total = N_SLOTS * VAL_DIM + N_SLOTS;
    for (int i = blockIdx.x * blockDim.x + threadIdx.x; i < total;
         i += gridDim.x * blockDim.x) {
        if (i < N_SLOTS * VAL_DIM) sgrads[i] = 0.f;
        else                       counts[i - N_SLOTS * VAL_DIM] = 0.f;
    }
}

// ---------------------------------------------------------------------------
// Kernel 5: scatter grads + counts into slots (float atomics)
// ---------------------------------------------------------------------------
__global__ void scatter_grads(const int* __restrict__ fidx, const float* __restrict__ grads,
                              float* __restrict__ sgrads, float* __restrict__ counts)
{
    int gid = blockIdx.x * blockDim.x + threadIdx.x;   // over N_QUERIES*TOPK*VAL_DIM
    int d = gid & (VAL_DIM - 1);
    int p = gid >> 6;
    int s = fidx[p];
    atomicAdd(&sgrads[(size_t)s * VAL_DIM + d], grads[gid]);
    if (d == 0) atomicAdd(&counts[s], 1.0f);
}

// ---------------------------------------------------------------------------
// Kernel 6: momentum + weight-decay update (memory-bound elementwise)
// ---------------------------------------------------------------------------
__global__ void update_vals(const float* __restrict__ vals, const float* __restrict__ mom,
                            const float* __restrict__ sgrads, const float* __restrict__ counts,
                            float* __restrict__ new_vals, float* __restrict__ new_mom)
{
    int gid = blockIdx.x * blockDim.x + threadIdx.x;   // over N_SLOTS*VAL_DIM
    int s = gid >> 6;
    float cnt = counts[s];
    float g = sgrads[gid];
    if (cnt > 0.f) g /= cnt;                 // cnt >= 1 when touched
    float v  = vals[gid];
    float nm = MOM * mom[gid] + g;
    float delta = -LR * (nm + WD * v);
    new_vals[gid] = v + (cnt > 0.f ? delta : 0.f);
    new_mom[gid]  = nm;
}

// ---------------------------------------------------------------------------
extern "C" void kernel_launch(void* const* d_in, const int* in_sizes, int n_in,
                              void* d_out, int out_size, void* d_ws, size_t ws_size,
                              hipStream_t stream)
{
    const float* Q    = (const float*)d_in[0];  // [8192,64]
    const float* Gr   = (const float*)d_in[1];  // [8192,4,64]
    const float* Keys = (const float*)d_in[2];  // [65536,64]
    const float* Vals = (const float*)d_in[3];  // [65536,64]
    const float* Mm   = (const float*)d_in[4];  // [65536,64]
    (void)in_sizes; (void)n_in; (void)out_size; (void)ws_size;

    float* out      = (float*)d_out;
    float* out_retr = out;                                             // 8192*4*64
    float* out_idx  = out_retr + (size_t)N_QUERIES * TOPK * VAL_DIM;   // 8192*4
    float* out_vals = out_idx  + (size_t)N_QUERIES * TOPK;             // 65536*64
    float* out_mom  = out_vals + (size_t)N_SLOTS * VAL_DIM;            // 65536*64

    char* ws = (char*)d_ws;
    size_t off = 0;
    float* pvals  = (float*)(ws + off); off += (size_t)N_QUERIES * NCHUNK * TOPK * 4;
    int*   pidx   = (int*)  (ws + off); off += (size_t)N_QUERIES * NCHUNK * TOPK * 4;
    int*   fidx   = (int*)  (ws + off); off += (size_t)N_QUERIES * TOPK * 4;
    float* counts = (float*)(ws + off); off += (size_t)N_SLOTS * 4;
    float* sgrads = (float*)(ws + off); off += (size_t)N_SLOTS * VAL_DIM * 4;

    dim3 gA(N_QUERIES / QT_WG, NCHUNK);                 // 128 x 8 workgroups
    topk_gemm<<<gA, 128, 0, stream>>>(Q, Keys, pvals, pidx);
    merge_topk<<<(N_QUERIES + 255) / 256, 256, 0, stream>>>(pvals, pidx, out_idx, fidx);
    gather_vals<<<(N_QUERIES * TOPK * VAL_DIM) / 256, 256, 0, stream>>>(fidx, Vals, out_retr);
    zero_ws<<<4096, 256, 0, stream>>>(sgrads, counts);
    scatter_grads<<<(N_QUERIES * TOPK * VAL_DIM) / 256, 256, 0, stream>>>(fidx, Gr, sgrads, counts);
    update_vals<<<(N_SLOTS * VAL_DIM) / 256, 256, 0, stream>>>(Vals, Mm, sgrads, counts, out_vals, out_mom);
}